// HashGrid_56573309223064
// MI455X (gfx1250) — compile-verified
//
#include <hip/hip_runtime.h>
#include <stdint.h>

typedef float v2f __attribute__((ext_vector_type(2)));

namespace {
constexpr int kLevels      = 16;
constexpr int kPtsPerBlock = 16;
constexpr int kThreads     = kLevels * kPtsPerBlock; // 256 = 8 waves (wave32)
constexpr uint32_t kP1 = 2654435761u;
constexpr uint32_t kP2 = 805459861u;
}

#if __has_builtin(__builtin_amdgcn_global_load_async_to_lds_b32)
#define HG_HAVE_ASYNC 1
#endif

__global__ __launch_bounds__(kThreads) void hashgrid_fwd(
    const float* __restrict__ inputs,  // [batch, 3] f32
    const v2f*   __restrict__ emb,     // [rows] (2 x f32 per row)
    v2f*         __restrict__ out,     // [batch, 16] float2  (== [batch,32] f32)
    int batch)
{
    __shared__ float sxyz[kPtsPerBlock * 3];

    const int t     = threadIdx.x;
    const int lvl   = t & (kLevels - 1);
    const int pib   = t >> 4;                        // point-in-block 0..15
    const int pbase = (int)blockIdx.x * kPtsPerBlock;

    int rem = batch - pbase;
    const int vp = rem < kPtsPerBlock ? (rem < 0 ? 0 : rem) : kPtsPerBlock;
    const int dwords = vp * 3;                       // <= 48 dwords to stage

    // ---- Stage the block's input coords into LDS (CDNA5 async path) ----
#ifdef HG_HAVE_ASYNC
    if (t < dwords) {
        typedef __attribute__((address_space(1))) int gint_t;
        typedef __attribute__((address_space(3))) int lint_t;
        __builtin_amdgcn_global_load_async_to_lds_b32(
            (gint_t*)(inputs + pbase * 3 + t),
            (lint_t*)&sxyz[t],
            /*offset=*/0, /*cpol=*/0);
    }
#if __has_builtin(__builtin_amdgcn_s_wait_asynccnt)
    __builtin_amdgcn_s_wait_asynccnt(0);
#else
    asm volatile("s_wait_asynccnt 0" ::: "memory");
#endif
    __syncthreads();
#else
    if (t < dwords) sxyz[t] = inputs[pbase * 3 + t];
    __syncthreads();
#endif

    if (pib >= vp) return;

    const float in0 = sxyz[pib * 3 + 0];
    const float in1 = sxyz[pib * 3 + 1];
    const float in2 = sxyz[pib * 3 + 2];

    // ---- Per-level grid coords ----
    const float scale = (float)(16 << lvl);          // res = 16 * 2^lvl
    const float x0 = in0 * scale, x1 = in1 * scale, x2 = in2 * scale;
    const float f0 = floorf(x0), f1 = floorf(x1), f2 = floorf(x2);
    const uint32_t i0 = (uint32_t)f0, i1 = (uint32_t)f1, i2 = (uint32_t)f2;
    const float xf0 = x0 - f0, xf1 = x1 - f1, xf2 = x2 - f2;

    // Level table size is 2^min(3*(4+lvl), 19); rounding-to-32 is a no-op on pow2.
    uint32_t shift = 3u * (4u + (uint32_t)lvl);
    if (shift > 19u) shift = 19u;
    const uint32_t mask = (1u << shift) - 1u;
    // Row offsets: prefix sums {0, 4096, 36864, 299008, +524288 ...}
    uint32_t off;
    if (lvl >= 3)      off = 299008u + ((uint32_t)lvl - 3u) * 524288u;
    else if (lvl == 2) off = 36864u;
    else if (lvl == 1) off = 4096u;
    else               off = 0u;

    // ---- Precompute the two hash contributions & weights per dimension ----
    const uint32_t h0a = i0,             h0b = i0 + 1u;
    const uint32_t h1a = i1 * kP1,       h1b = (i1 + 1u) * kP1;
    const uint32_t h2a = i2 * kP2,       h2b = (i2 + 1u) * kP2;
    const float w0a = 1.0f - xf0, w0b = xf0;
    const float w1a = 1.0f - xf1, w1b = xf1;
    const float w2a = 1.0f - xf2, w2b = xf2;

    // ---- 8 corner gathers (unrolled so all 8 b64 loads issue back-to-back) ----
    float acc0 = 0.0f, acc1 = 0.0f;
#pragma unroll
    for (int n = 0; n < 8; ++n) {
        const uint32_t h = ((n & 1) ? h0b : h0a) ^
                           ((n & 2) ? h1b : h1a) ^
                           ((n & 4) ? h2b : h2a);
        const uint32_t idx = (h & mask) + off;
        const v2f f = emb[idx];                      // global_load_b64, L2-resident
        const float w = ((n & 1) ? w0b : w0a) *
                        ((n & 2) ? w1b : w1a) *
                        ((n & 4) ? w2b : w2a);
        acc0 = fmaf(w, f.x, acc0);
        acc1 = fmaf(w, f.y, acc1);
    }

    // Coalesced: consecutive threads -> consecutive float2. Nontemporal so the
    // 256 MiB output stream doesn't evict the 57 MB embedding table from L2.
    v2f r; r.x = acc0; r.y = acc1;
    __builtin_nontemporal_store(r, out + (size_t)pbase * kLevels + t);
}

extern "C" void kernel_launch(void* const* d_in, const int* in_sizes, int n_in,
                              void* d_out, int out_size, void* d_ws, size_t ws_size,
                              hipStream_t stream) {
    const float* inputs = (const float*)d_in[0];
    const v2f*   emb    = (const v2f*)d_in[1];
    v2f*         out    = (v2f*)d_out;
    const int batch  = in_sizes[0] / 3;
    const int blocks = (batch + kPtsPerBlock - 1) / kPtsPerBlock;
    hashgrid_fwd<<<blocks, kThreads, 0, stream>>>(inputs, emb, out, batch);
}